// ClusteringLayer_26749056319813
// MI455X (gfx1250) — compile-verified
//
#include <hip/hip_runtime.h>
#include <cstdint>
#include <cstddef>

#define THR 0.05f
#define WAVES_PER_BLOCK 8
#define BLOCK_THREADS (WAVES_PER_BLOCK * 32)

// Broadcast the value held in `lane` of v across the wave (uniform lane index).
__device__ __forceinline__ float bcast(float v, int lane) {
    return __builtin_bit_cast(
        float, __builtin_amdgcn_readlane(__builtin_bit_cast(int, v), lane));
}

__device__ __forceinline__ void async_load_b64(uint32_t lds_off, const void* gaddr) {
    // CDNA5 async copy: 8 bytes per lane, global -> LDS, tracked by ASYNCcnt.
    asm volatile("global_load_async_to_lds_b64 %0, %1, off"
                 :: "v"(lds_off), "v"(gaddr)
                 : "memory");
}

__global__ void __launch_bounds__(BLOCK_THREADS)
cluster_kernel(const float* __restrict__ x, float* __restrict__ out, int n_rows) {
    __shared__ float sh[WAVES_PER_BLOCK][2][64];

    const int lane = threadIdx.x & 31;
    const int wave = threadIdx.x >> 5;
    const int waves_total = gridDim.x * WAVES_PER_BLOCK;
    int row = blockIdx.x * WAVES_PER_BLOCK + wave;

    if (row >= n_rows) return;  // uniform per wave

    // Per-wave double buffers; low 32 bits of a __shared__ flat address are the
    // workgroup-relative LDS byte offset (shared aperture).
    const uint32_t lds0 = (uint32_t)(uintptr_t)&sh[wave][0][lane * 2];
    const uint32_t lds1 = (uint32_t)(uintptr_t)&sh[wave][1][lane * 2];

    // Prefetch first line: lane L copies bytes [8L, 8L+8) of the 256B line.
    async_load_b64(lds0, x + (size_t)row * 64 + lane * 2);

    int buf = 0;
    for (;;) {
        const int next = row + waves_total;
        const bool have_next = next < n_rows;  // uniform
        if (have_next) {
            async_load_b64(buf ? lds0 : lds1, x + (size_t)next * 64 + lane * 2);
            asm volatile("s_wait_asynccnt 1" ::: "memory");  // current line landed
        } else {
            asm volatile("s_wait_asynccnt 0" ::: "memory");
        }

        const float* cur = &sh[wave][buf][0];
        float v0 = cur[lane];        // element  lane      of the line
        float v1 = cur[lane + 32];   // element (lane+32)  of the line

        // clustered == "has been overwritten"; kept as divergent i1 -> wave masks
        bool cl0 = false, cl1 = false;

        // Phase 1: seeds 0..31 live in v0. For v1 (cols 32..63) col>idx always.
#pragma unroll
        for (int idx = 0; idx < 32; ++idx) {
            const float sel = bcast(v0, idx);
            const bool c0 = (__builtin_fabsf(v0 - sel) <= THR) && (lane > idx);
            const bool c1 = (__builtin_fabsf(v1 - sel) <= THR);
            if (c0 && !cl0) v0 = sel;
            if (c1 && !cl1) v1 = sel;
            cl0 = cl0 || c0;
            cl1 = cl1 || c1;
        }
        // Phase 2: seeds 32..63 live in v1; v0 can no longer change (col<=31).
#pragma unroll
        for (int k = 0; k < 32; ++k) {
            const float sel = bcast(v1, k);
            const bool c1 = (__builtin_fabsf(v1 - sel) <= THR) && (lane > k);
            if (c1 && !cl1) v1 = sel;
            cl1 = cl1 || c1;
        }

        float* o = out + (size_t)row * 64;
        o[lane]      = v0;
        o[lane + 32] = v1;

        if (!have_next) break;
        row = next;
        buf ^= 1;
    }
}

extern "C" void kernel_launch(void* const* d_in, const int* in_sizes, int n_in,
                              void* d_out, int out_size, void* d_ws, size_t ws_size,
                              hipStream_t stream) {
    (void)n_in; (void)out_size; (void)d_ws; (void)ws_size;
    const float* x = (const float*)d_in[0];
    float* out = (float*)d_out;
    const int n_rows = in_sizes[0] / 64;  // 64 consecutive floats per cacheline
    int blocks = (n_rows + WAVES_PER_BLOCK - 1) / WAVES_PER_BLOCK;
    if (blocks > 8192) blocks = 8192;
    if (blocks < 1) blocks = 1;
    hipLaunchKernelGGL(cluster_kernel, dim3(blocks), dim3(BLOCK_THREADS), 0, stream,
                       x, out, n_rows);
}